// SKANLinear_random_59313498358102
// MI455X (gfx1250) — compile-verified
//
#include <hip/hip_runtime.h>

// SKAN linear: y[b,o] = sum_i weight[o,i] * sin(w[o,i] * x_ext[b,i]),
// x_ext = [x | 1].  B=2048, IN=256, OUT=256, K=IN+1=257.
//
// TRANS-pipe bound (134.7M v_sin_f32).  Not WMMA-expressible: sin() is applied
// per (b,o,i) before the K-reduction, so no A[m,k]*B[k,n] factorization exists.
// CDNA5 strategy: async-to-LDS staging of x tiles (ASYNCcnt path), scalar-path
// (SMEM/SALU-float) handling of wave-uniform w/weight rows including the
// 1/(2*pi) range reduction for v_sin_f32, wave32 lane=b mapping, and an
// LDS transpose for coalesced stores.
//
// Inner loop per element: v_mul (uniform-scaled w * x), v_sin_f32 (TRANS),
// v_fmac -- 2 VALU + 1 TRANS, with the range-reduction multiply retired once
// per (o,i) on the scalar unit instead of per lane.

#define B_GLOBAL 2048
#define IN_DIM   256
#define OUT_DIM  256
#define KCOL     257      // IN+1 (bias column folded analytically)

#define B_TILE     32     // batch rows per block (== one lane per row, wave32)
#define O_TILE     64     // output cols per block
#define THREADS    256    // 8 waves
#define O_PER_WAVE 8      // O_TILE / 8 waves
#define XPITCH     260    // LDS row pitch (floats): 16B-aligned rows for b128 async writes
#define YPITCH     65     // odd pitch -> conflict-free transpose through LDS

#define INV_2PI  0.15915494309189535f   // v_sin_f32 takes revolutions

__global__ __launch_bounds__(THREADS)
void skan_sin_kernel(const float* __restrict__ x,       // [2048][256]
                     const float* __restrict__ weight,  // [256][257]
                     const float* __restrict__ w,       // [256][257]
                     float* __restrict__ y)             // [2048][256]
{
    __shared__ __align__(16) float xs[B_TILE * XPITCH];   // ~33 KB
    __shared__ float ytile[B_TILE * YPITCH];              // ~8 KB

    const int tid       = threadIdx.x;
    const int bBase     = blockIdx.y * B_TILE;
    const int oTileBase = blockIdx.x * O_TILE;

    // ------------------------------------------------------------------
    // Stage x tile (32 rows x 256 floats) into LDS with CDNA5 async copies.
    // 2048 16-byte chunks, 8 per thread; consecutive tids hit consecutive
    // global addresses (fully coalesced).  GVS addressing: per-lane LDS
    // offset VGPR, per-lane global offset VGPR, SGPR-pair base.
    // ------------------------------------------------------------------
    const float* xg = x + (size_t)bBase * IN_DIM;
    // Low 32 bits of a generic LDS pointer are the LDS byte offset.
    unsigned xsBase = (unsigned)(size_t)(&xs[0]);
    #pragma unroll
    for (int k = 0; k < 8; ++k) {
        int c    = tid + k * THREADS;     // chunk id 0..2047
        int row  = c >> 6;                // 64 float4-chunks per row
        int quad = c & 63;
        unsigned ldsOff = xsBase + (unsigned)((row * XPITCH + quad * 4) * 4);
        unsigned gOff   = (unsigned)((row * IN_DIM + quad * 4) * 4);
        asm volatile("global_load_async_to_lds_b128 %0, %1, %2"
                     :: "v"(ldsOff), "v"(gOff), "s"(xg)
                     : "memory");
    }
    asm volatile("s_wait_asynccnt 0" ::: "memory");
    __syncthreads();

    // ------------------------------------------------------------------
    // Compute: lane = batch row within tile, wave owns 8 output columns.
    // w/weight accesses are wave-uniform -> scalar (SMEM/K$) loads, and the
    // 1/(2*pi) pre-scale of w stays on the scalar unit (uniform value).
    // ------------------------------------------------------------------
    const int lane  = tid & 31;
    const int wave  = __builtin_amdgcn_readfirstlane(tid >> 5);  // force SGPR
    const int oBase = oTileBase + wave * O_PER_WAVE;

    const float* xrow = &xs[lane * XPITCH];

    float acc[O_PER_WAVE];
    #pragma unroll
    for (int j = 0; j < O_PER_WAVE; ++j) {
        const int o = oBase + j;                       // wave-uniform
        // bias column: x_ext[b,256] == 1
        const float wb = w[(size_t)o * KCOL + IN_DIM] * INV_2PI;  // uniform
        acc[j] = weight[(size_t)o * KCOL + IN_DIM] *
                 __builtin_amdgcn_sinf(wb);
    }

    // Inner loop per i: 1 ds_load_b32 + 8 x (v_mul, v_sin_f32, v_fmac).
    #pragma unroll 4
    for (int i = 0; i < IN_DIM; ++i) {
        const float xv = xrow[i];
        #pragma unroll
        for (int j = 0; j < O_PER_WAVE; ++j) {
            const int o = oBase + j;                          // wave-uniform
            const float wvs = w[(size_t)o * KCOL + i] * INV_2PI; // s_load + s_mul_f32
            const float gv  = weight[(size_t)o * KCOL + i];      // s_load
            acc[j] += gv * __builtin_amdgcn_sinf(wvs * xv);
        }
    }

    // ------------------------------------------------------------------
    // Transpose through LDS so the global store is coalesced.
    // ------------------------------------------------------------------
    #pragma unroll
    for (int j = 0; j < O_PER_WAVE; ++j) {
        ytile[lane * YPITCH + wave * O_PER_WAVE + j] = acc[j];
    }
    __syncthreads();

    #pragma unroll
    for (int k = 0; k < 8; ++k) {
        int idx = tid + k * THREADS;       // 0..2047
        int bl  = idx >> 6;                // / O_TILE
        int ol  = idx & (O_TILE - 1);
        y[(size_t)(bBase + bl) * OUT_DIM + oTileBase + ol] =
            ytile[bl * YPITCH + ol];
    }
}

extern "C" void kernel_launch(void* const* d_in, const int* in_sizes, int n_in,
                              void* d_out, int out_size, void* d_ws, size_t ws_size,
                              hipStream_t stream) {
    (void)in_sizes; (void)n_in; (void)out_size; (void)d_ws; (void)ws_size;
    const float* x      = (const float*)d_in[0];   // [2048,256]
    const float* weight = (const float*)d_in[1];   // [256,257]
    const float* w      = (const float*)d_in[2];   // [256,257]
    float* out          = (float*)d_out;           // [2048,256]

    dim3 grid(OUT_DIM / O_TILE, B_GLOBAL / B_TILE);  // (4, 64) = 256 blocks
    skan_sin_kernel<<<grid, dim3(THREADS), 0, stream>>>(x, weight, w, out);
}